// TwinningEdgeAttention_33663953666650
// MI455X (gfx1250) — compile-verified
//
#include <hip/hip_runtime.h>

// Problem constants (fixed by the reference).
#define E_EDGES 640000
#define DE      128
#define DN      128
#define NNODES  50000

#define SA 260   // LDS stride for combined[16][256] tile (conflict-free)
#define SH 132   // LDS stride for h[16][128] tile (conflict-free)

typedef __attribute__((ext_vector_type(2))) float v2f;
typedef __attribute__((ext_vector_type(8))) float v8f;

// ---------------------------------------------------------------------------
// Kernel 1: scatter edge features into per-node src/dst sums + counts.
// One thread per (edge, feature). Uses native GLOBAL_ATOMIC_ADD_F32.
// ---------------------------------------------------------------------------
__global__ void __launch_bounds__(256)
edge_scatter_kernel(const float* __restrict__ ef,
                    const long long* __restrict__ eidx,
                    float* __restrict__ sum_src,
                    float* __restrict__ sum_dst,
                    float* __restrict__ cnt_src,
                    float* __restrict__ cnt_dst) {
  unsigned gid = blockIdx.x * 256u + threadIdx.x;
  unsigned e = gid >> 7;       // edge id
  unsigned f = gid & 127u;     // feature id
  if (e >= E_EDGES) return;

  float v = ef[(size_t)e * DE + f];
  int s = (int)eidx[e];
  int d = (int)eidx[E_EDGES + e];
  atomicAdd(&sum_src[(size_t)s * DE + f], v);
  atomicAdd(&sum_dst[(size_t)d * DE + f], v);
  if (f == 0) {
    atomicAdd(&cnt_src[s], 1.0f);
    atomicAdd(&cnt_dst[d], 1.0f);
  }
}

// ---------------------------------------------------------------------------
// Kernel 2: fused normalize + GEMM1(relu) + GEMM2(sigmoid) via f32 WMMA.
// One block = 16 node-rows, 8 waves; each wave owns a 16-wide column tile.
// ---------------------------------------------------------------------------
__global__ void __launch_bounds__(256)
mlp_wmma_kernel(const float* __restrict__ sum_src,
                const float* __restrict__ sum_dst,
                const float* __restrict__ cnt_src,
                const float* __restrict__ cnt_dst,
                const float* __restrict__ W1,   // [256][128] row-major
                const float* __restrict__ b1,   // [128]
                const float* __restrict__ W2,   // [128][128] row-major
                const float* __restrict__ b2,   // [128]
                float* __restrict__ out) {      // [N][128]
  __shared__ float comb[16 * SA];
  __shared__ float hbuf[16 * SH];

  const int row0 = blockIdx.x * 16;
  const int tid  = threadIdx.x;

  // --- Stage normalized combined tile [16 rows x 256 cols] into LDS ---
  for (int i = 0; i < 16; ++i) {
    int idx = tid + i * 256;
    int r = idx >> 8;        // 0..15
    int c = idx & 255;       // 0..255
    int node = row0 + r;
    float v;
    if (c < DE) {
      v = sum_src[(size_t)node * DE + c] / fmaxf(cnt_src[node], 1.0f);
    } else {
      v = sum_dst[(size_t)node * DE + (c - DE)] / fmaxf(cnt_dst[node], 1.0f);
    }
    comb[r * SA + c] = v;
  }
  __syncthreads();

  const int lane = tid & 31;
  const int wave = tid >> 5;     // 0..7 : column tile id
  const int lo   = lane & 15;    // row (A) / col (B,C,D) within tile
  const int hi   = lane >> 4;    // selects K pair (A/B) or M+8 (C/D)
  const int n0   = wave * 16;

  // --- GEMM1: h = relu(comb[16,256] @ W1[256,128] + b1), K-steps of 4 ---
  v8f acc = {};
  for (int k = 0; k < 2 * DE; k += 4) {
    int ka = k + 2 * hi;
    v2f a, b;
    a.x = comb[lo * SA + ka];
    a.y = comb[lo * SA + ka + 1];
    b.x = W1[ka * DE + n0 + lo];
    b.y = W1[(ka + 1) * DE + n0 + lo];
    acc = __builtin_amdgcn_wmma_f32_16x16x4_f32(
        false, a, false, b, (short)0, acc, false, false);
  }
  {
    float bias = b1[n0 + lo];
    #pragma unroll
    for (int v = 0; v < 8; ++v) {
      float x = acc[v] + bias;
      hbuf[(v + 8 * hi) * SH + n0 + lo] = fmaxf(x, 0.0f);
    }
  }
  __syncthreads();

  // --- GEMM2: out = sigmoid(h[16,128] @ W2[128,128] + b2) ---
  v8f acc2 = {};
  for (int k = 0; k < DE; k += 4) {
    int ka = k + 2 * hi;
    v2f a, b;
    a.x = hbuf[lo * SH + ka];
    a.y = hbuf[lo * SH + ka + 1];
    b.x = W2[ka * DN + n0 + lo];
    b.y = W2[(ka + 1) * DN + n0 + lo];
    acc2 = __builtin_amdgcn_wmma_f32_16x16x4_f32(
        false, a, false, b, (short)0, acc2, false, false);
  }
  {
    float bias = b2[n0 + lo];
    #pragma unroll
    for (int v = 0; v < 8; ++v) {
      float x = acc2[v] + bias;
      x = 1.0f / (1.0f + __expf(-x));
      out[(size_t)(row0 + v + 8 * hi) * DN + n0 + lo] = x;
    }
  }
}

// ---------------------------------------------------------------------------
extern "C" void kernel_launch(void* const* d_in, const int* in_sizes, int n_in,
                              void* d_out, int out_size, void* d_ws, size_t ws_size,
                              hipStream_t stream) {
  const float*     ef   = (const float*)d_in[0];      // [E,128] f32
  const float*     W1   = (const float*)d_in[1];      // [256,128] f32
  const float*     b1   = (const float*)d_in[2];      // [128] f32
  const float*     W2   = (const float*)d_in[3];      // [128,128] f32
  const float*     b2   = (const float*)d_in[4];      // [128] f32
  const long long* eidx = (const long long*)d_in[5];  // [2,E] int64
  // d_in[6] = num_nodes scalar (compile-time constant here)

  float* sum_src = (float*)d_ws;
  float* sum_dst = sum_src + (size_t)NNODES * DE;
  float* cnt_src = sum_dst + (size_t)NNODES * DE;
  float* cnt_dst = cnt_src + NNODES;

  size_t zero_bytes = ((size_t)2 * NNODES * DE + 2 * (size_t)NNODES) * sizeof(float);
  hipMemsetAsync(d_ws, 0, zero_bytes, stream);

  unsigned total_threads = (unsigned)E_EDGES * DE;  // 81,920,000
  edge_scatter_kernel<<<total_threads / 256, 256, 0, stream>>>(
      ef, eidx, sum_src, sum_dst, cnt_src, cnt_dst);

  mlp_wmma_kernel<<<NNODES / 16, 256, 0, stream>>>(
      sum_src, sum_dst, cnt_src, cnt_dst, W1, b1, W2, b2, (float*)d_out);
}